// DeepGATEncoder_42382737277575
// MI455X (gfx1250) — compile-verified
//
#include <hip/hip_runtime.h>
#include <hip/hip_bf16.h>
#include <math.h>

// ---------------- problem constants (from reference) ----------------
#define N_NODES   3072
#define N_EDGES   98304
#define F_IN_DIM  300
#define HID_DIM   300     // per-head out dim, layers 0/1
#define HIDP      320     // padded
#define FEAT      3000    // HEADS*HID
#define FEATP     3200    // padded (10 * 320)
#define OUT_ATT   600
#define OUTP      640     // padded
#define HEADS     10
#define N_GRAPHS  96
#define G_PAD     128
#define MLP_HID   600
#define MLPP      640
#define NOUT      768
#define ALPHA_LRELU 0.02f
#define NEG_INF   (-9e15f)

typedef float v2f __attribute__((ext_vector_type(2)));
typedef float v8f __attribute__((ext_vector_type(8)));

// Full-precision f32 tensor op: D(16x16) = A(16x4) x B(4x16) + C
__device__ __forceinline__ v8f wmma4(v2f a, v2f b, v8f c) {
  return __builtin_amdgcn_wmma_f32_16x16x4_f32(false, a, false, b, (short)0, c,
                                               false, false);
}

__device__ __forceinline__ float lrelu(float v) {
  return v > 0.f ? v : ALPHA_LRELU * v;
}

// ---------------- mask construction ----------------
__global__ void k_clear_bytes(unsigned char* p, long n) {
  long i = (long)blockIdx.x * blockDim.x + threadIdx.x;
  if (i < n) p[i] = 0;
}

__global__ void k_scatter_edges(const int* __restrict__ ei,
                                unsigned char* __restrict__ mask, int E, int Nn) {
  int e = blockIdx.x * blockDim.x + threadIdx.x;
  if (e < E) {
    int s = ei[e];        // edge_index[0][e]
    int t = ei[E + e];    // edge_index[1][e]
    mask[(long)s * Nn + t] = 1;
  }
}

// ---------------- weight staging with zero padding ----------------
// dst[h][r][c] (r<dstR, c<dstC, ld=ldd): r decomposes as blk*rowBlkD + ro;
// src row = blk*rowBlkS + ro.  Zero where ro>=rowBlkS or c>=srcCols.
__global__ void k_stage_pad(const float* __restrict__ src, float* __restrict__ dst,
                            int dstR, int dstC, int ldd,
                            int rowBlkS, int rowBlkD, int srcCols, int lds,
                            long srcHS, long dstHS) {
  int idx = blockIdx.x * blockDim.x + threadIdx.x;
  int h = blockIdx.y;
  if (idx >= dstR * dstC) return;
  int r = idx / dstC, c = idx % dstC;
  int blk = r / rowBlkD, ro = r % rowBlkD;
  float v = 0.f;
  if (ro < rowBlkS && c < srcCols)
    v = src[(long)h * srcHS + (long)(blk * rowBlkS + ro) * lds + c];
  dst[(long)h * dstHS + (long)r * ldd + c] = v;
}

// ---------------- guard-free WMMA-f32 GEMM on padded operands ----------------
// block = 128 (4 waves); wave computes 16 rows x 64 cols (4 accumulators).
// Requires: rows covered by grid all valid, cols padded, K % 4 == 0.
// epilogue: 0 none, 1 bias+relu, 2 bias
__global__ void k_wmma_gemm(const float* __restrict__ A,
                            const float* __restrict__ B,
                            float* __restrict__ C,
                            int K, int lda, int ldb, int ldc,
                            long sA, long sB, long sC,
                            const float* __restrict__ bias, int epilogue) {
  const int lane  = threadIdx.x & 31;
  const int wave  = threadIdx.x >> 5;
  const int m16   = lane & 15;
  const int khalf = (lane >> 4) << 1;          // 0 | 2
  A += (long)blockIdx.z * sA;
  B += (long)blockIdx.z * sB;
  C += (long)blockIdx.z * sC;
  const int rowTile = blockIdx.y * 64 + wave * 16;
  const int colBase = blockIdx.x * 64;

  const float* Ap = A + (long)(rowTile + m16) * lda + khalf;
  const float* Bp = B + (long)khalf * ldb + colBase + m16;
  const long bstep = (long)4 * ldb;

  v8f a0 = {}, a1 = {}, a2 = {}, a3 = {};
  for (int k = 0; k < K; k += 4) {
    v2f a = *(const v2f*)Ap;                 // K-pair, 8B aligned
    v2f b0, b1, b2, b3;
    b0.x = Bp[0];        b0.y = Bp[ldb];
    b1.x = Bp[16];       b1.y = Bp[ldb + 16];
    b2.x = Bp[32];       b2.y = Bp[ldb + 32];
    b3.x = Bp[48];       b3.y = Bp[ldb + 48];
    a0 = wmma4(a, b0, a0);
    a1 = wmma4(a, b1, a1);
    a2 = wmma4(a, b2, a2);
    a3 = wmma4(a, b3, a3);
    Ap += 4;
    Bp += bstep;
  }

  // D layout: lane r holds row rbase+r; lanes>=16 offset by 8
  const int rbase = rowTile + ((lane >> 4) << 3);
  float* Cp = C + (long)rbase * ldc + colBase + m16;
  v8f acc[4] = {a0, a1, a2, a3};
#pragma unroll
  for (int t = 0; t < 4; ++t) {
    float bv = (epilogue != 0) ? bias[colBase + t * 16 + m16] : 0.f;
#pragma unroll
    for (int r = 0; r < 8; ++r) {
      float v = acc[t][r];
      if (epilogue == 1)      v = fmaxf(v + bv, 0.f);
      else if (epilogue == 2) v = v + bv;
      Cp[(long)r * ldc + t * 16] = v;
    }
  }
}

// ---------------- attention score projections s,d ----------------
__global__ void k_compute_sd(const float* __restrict__ Wh,
                             const float* __restrict__ a_src,
                             const float* __restrict__ a_dst,
                             float* __restrict__ s, float* __restrict__ d,
                             int Nn, int H, int O, int ldw, int headOff) {
  int idx = blockIdx.x * blockDim.x + threadIdx.x;
  if (idx >= Nn * H) return;
  int n = idx % Nn, h = idx / Nn;
  const float* w  = Wh + (long)n * ldw + (long)h * headOff;
  const float* as = a_src + (long)h * O;
  const float* ad = a_dst + (long)h * O;
  float ss = 0.f, dd = 0.f;
  for (int o = 0; o < O; ++o) {
    float v = w[o];
    ss = fmaf(v, as[o], ss);
    dd = fmaf(v, ad[o], dd);
  }
  s[(long)h * Nn + n] = ss;
  d[(long)h * Nn + n] = dd;
}

// ---------------- softmax row statistics (online, wave32) ----------------
__global__ void k_gat_stats(const unsigned char* __restrict__ mask,
                            const float* __restrict__ s,
                            const float* __restrict__ dv,
                            float* __restrict__ rmax, float* __restrict__ rinv,
                            int Nn) {
  const int i = blockIdx.x, h = blockIdx.y, lane = threadIdx.x;
  const float sI = s[(long)h * Nn + i];
  const float* dh = dv + (long)h * Nn;
  const unsigned char* mrow = mask + (long)i * Nn;
  float m = -INFINITY, sum = 0.f;
  for (int j = lane; j < Nn; j += 32) {
    float e = mrow[j] ? lrelu(sI + dh[j]) : NEG_INF;  // finite NEG_INF == jnp.where
    if (e > m) { sum = sum * expf(m - e) + 1.f; m = e; }
    else       { sum += expf(e - m); }
  }
  for (int off = 16; off > 0; off >>= 1) {
    float m2 = __shfl_xor(m, off, 32);
    float s2 = __shfl_xor(sum, off, 32);
    float M = fmaxf(m, m2);
    sum = sum * expf(m - M) + s2 * expf(m2 - M);
    m = M;
  }
  if (lane == 0) {
    rmax[(long)h * Nn + i] = m;
    rinv[(long)h * Nn + i] = 1.f / sum;
  }
}

// ---------------- fused attention apply: out = ELU(softmax(e) @ Wh) ----------
// P fragments synthesized in registers (flash style); guard-free on padded Wh.
__global__ void k_gat_apply(const float* __restrict__ Wh,
                            const unsigned char* __restrict__ mask,
                            const float* __restrict__ s,
                            const float* __restrict__ dv,
                            const float* __restrict__ rmax,
                            const float* __restrict__ rinv,
                            float* __restrict__ out,
                            int Nn, int ldw, int ldo, long headStride) {
  const int lane  = threadIdx.x & 31;
  const int wave  = threadIdx.x >> 5;
  const int m16   = lane & 15;
  const int khalf = (lane >> 4) << 1;
  const int h = blockIdx.z;

  const float* WhH = Wh + (long)h * headStride;
  float*       oH  = out + (long)h * headStride;
  const float* sh  = s    + (long)h * Nn;
  const float* dh  = dv   + (long)h * Nn;

  const int rowTile = (blockIdx.y * 4 + wave) * 16;
  const int colBase = blockIdx.x * 64;
  const int row = rowTile + m16;
  const float sI = sh[row];
  const float mI = rmax[(long)h * Nn + row];
  const float rI = rinv[(long)h * Nn + row];
  const unsigned char* mrow = mask + (long)row * Nn + khalf;
  const float* dp = dh + khalf;
  const float* Bp = WhH + (long)khalf * ldw + colBase + m16;
  const long bstep = (long)4 * ldw;

  v8f a0 = {}, a1 = {}, a2 = {}, a3 = {};
  for (int j = 0; j < Nn; j += 4) {
    v2f dpair = *(const v2f*)dp;             // d[j0], d[j0+1] (8B aligned)
    uchar2 mm = *(const uchar2*)mrow;        // mask bytes (2B aligned)
    float e0 = mm.x ? lrelu(sI + dpair.x) : NEG_INF;
    float e1 = mm.y ? lrelu(sI + dpair.y) : NEG_INF;
    v2f a;
    a.x = expf(e0 - mI) * rI;   // masked entries underflow to 0; empty rows -> 1/N
    a.y = expf(e1 - mI) * rI;
    v2f b0, b1, b2, b3;
    b0.x = Bp[0];        b0.y = Bp[ldw];
    b1.x = Bp[16];       b1.y = Bp[ldw + 16];
    b2.x = Bp[32];       b2.y = Bp[ldw + 32];
    b3.x = Bp[48];       b3.y = Bp[ldw + 48];
    a0 = wmma4(a, b0, a0);
    a1 = wmma4(a, b1, a1);
    a2 = wmma4(a, b2, a2);
    a3 = wmma4(a, b3, a3);
    dp += 4;
    mrow += 4;
    Bp += bstep;
  }

  const int rbase = rowTile + ((lane >> 4) << 3);
  float* Op = oH + (long)rbase * ldo + colBase + m16;
  v8f acc[4] = {a0, a1, a2, a3};
#pragma unroll
  for (int t = 0; t < 4; ++t) {
#pragma unroll
    for (int r = 0; r < 8; ++r) {
      float v = acc[t][r];
      v = v > 0.f ? v : expm1f(v);           // ELU (alpha=1); pads stay exactly 0
      Op[(long)r * ldo + t * 16] = v;
    }
  }
}

// ---------------- pooling / misc ----------------
__global__ void k_zero_f32(float* p, long n) {
  long i = (long)blockIdx.x * blockDim.x + threadIdx.x;
  if (i < n) p[i] = 0.f;
}

__global__ void k_pool_accum(const float* __restrict__ hf, const int* __restrict__ batch,
                             float* __restrict__ sums, float* __restrict__ cnt,
                             int Nn, int F, int ldh, int ldsum) {
  long idx = (long)blockIdx.x * blockDim.x + threadIdx.x;
  if (idx >= (long)Nn * F) return;
  int n = (int)(idx / F), f = (int)(idx % F);
  int g = batch[n];
  atomicAdd(&sums[(long)g * ldsum + f], hf[(long)n * ldh + f]);
  if (f == 0) atomicAdd(&cnt[g], 1.f);
}

__global__ void k_pool_norm(const float* __restrict__ sums, const float* __restrict__ cnt,
                            float* __restrict__ pooled, int G, int F, int ld) {
  int idx = blockIdx.x * blockDim.x + threadIdx.x;
  if (idx >= G * F) return;
  int g = idx / F, f = idx % F;
  pooled[(long)g * ld + f] = sums[(long)g * ld + f] / fmaxf(cnt[g], 1.f);
}

__global__ void k_copy_out(const float* __restrict__ src, float* __restrict__ dst,
                           int R, int Csz, int lds) {
  int idx = blockIdx.x * blockDim.x + threadIdx.x;
  if (idx >= R * Csz) return;
  dst[idx] = src[(long)(idx / Csz) * lds + (idx % Csz)];
}

// ---------------- host orchestration ----------------
static inline int ceil_div(int a, int b) { return (a + b - 1) / b; }

extern "C" void kernel_launch(void* const* d_in, const int* in_sizes, int n_in,
                              void* d_out, int out_size, void* d_ws, size_t ws_size,
                              hipStream_t stream) {
  const float* x      = (const float*)d_in[0];
  const int*   eidx   = (const int*)  d_in[1];
  const int*   batch  = (const int*)  d_in[2];
  const float* W0     = (const float*)d_in[3];
  const float* a0_src = (const float*)d_in[4];
  const float* a0_dst = (const float*)d_in[5];
  const float* W1     = (const float*)d_in[6];
  const float* a1_src = (const float*)d_in[7];
  const float* a1_dst = (const float*)d_in[8];
  const float* W_out  = (const float*)d_in[9];
  const float* ao_src = (const float*)d_in[10];
  const float* ao_dst = (const float*)d_in[11];
  const float* Wm1    = (const float*)d_in[12];
  const float* bm1    = (const float*)d_in[13];
  const float* Wm2    = (const float*)d_in[14];
  const float* bm2    = (const float*)d_in[15];
  float* out = (float*)d_out;
  (void)in_sizes; (void)n_in; (void)out_size; (void)ws_size;

  const int N = N_NODES, H = HEADS, G = N_GRAPHS;

  // ---- workspace carve-up (~186 MB total) ----
  size_t off = 0;
  auto carve = [&](size_t bytes) -> void* {
    void* p = (char*)d_ws + off;
    off += (bytes + 255) & ~(size_t)255;
    return p;
  };
  unsigned char* mask = (unsigned char*)carve((size_t)N * N);
  float* bufA  = (float*)carve((size_t)N * FEATP * 4);     // Wh scratch (padded)
  float* bufB  = (float*)carve((size_t)N * FEATP * 4);
  float* bufC  = (float*)carve((size_t)N * FEATP * 4);
  float* W0p   = (float*)carve((size_t)H * F_IN_DIM * HIDP * 4);   // [10][300][320]
  float* W1p   = (float*)carve((size_t)H * FEATP * HIDP * 4);      // [10][3200][320]
  float* Wop   = (float*)carve((size_t)FEATP * OUTP * 4);          // [3200][640]
  float* Wm1p  = (float*)carve((size_t)MLPP * MLPP * 4);           // [640][640]
  float* Wm2p  = (float*)carve((size_t)MLPP * NOUT * 4);           // [640][768]
  float* bm1p  = (float*)carve((size_t)MLPP * 4);
  float* bm2p  = (float*)carve((size_t)NOUT * 4);
  float* sbuf  = (float*)carve((size_t)H * N * 4);
  float* dbuf  = (float*)carve((size_t)H * N * 4);
  float* rmaxb = (float*)carve((size_t)H * N * 4);
  float* rinvb = (float*)carve((size_t)H * N * 4);
  float* psum  = (float*)carve((size_t)G_PAD * OUTP * 4);
  float* pcnt  = (float*)carve((size_t)G * 4);
  float* pooled= (float*)carve((size_t)G_PAD * OUTP * 4);
  float* hidden= (float*)carve((size_t)G_PAD * MLPP * 4);
  float* outp  = (float*)carve((size_t)G_PAD * NOUT * 4);

  // ---- adjacency mask ----
  {
    long nb = (long)N * N;
    k_clear_bytes<<<dim3((unsigned)((nb + 255) / 256)), 256, 0, stream>>>(mask, nb);
    k_scatter_edges<<<ceil_div(N_EDGES, 256), 256, 0, stream>>>(eidx, mask, N_EDGES, N);
  }

  // ---- stage weights into zero-padded layouts ----
  k_stage_pad<<<dim3(ceil_div(F_IN_DIM * HIDP, 256), H), 256, 0, stream>>>(
      W0, W0p, F_IN_DIM, HIDP, HIDP, F_IN_DIM, F_IN_DIM, HID_DIM, HID_DIM,
      (long)F_IN_DIM * HID_DIM, (long)F_IN_DIM * HIDP);
  k_stage_pad<<<dim3(ceil_div(FEATP * HIDP, 256), H), 256, 0, stream>>>(
      W1, W1p, FEATP, HIDP, HIDP, HID_DIM, HIDP, HID_DIM, HID_DIM,
      (long)FEAT * HID_DIM, (long)FEATP * HIDP);
  k_stage_pad<<<dim3(ceil_div(FEATP * OUTP, 256), 1), 256, 0, stream>>>(
      W_out, Wop, FEATP, OUTP, OUTP, HID_DIM, HIDP, OUT_ATT, OUT_ATT, 0L, 0L);
  k_stage_pad<<<dim3(ceil_div(MLPP * MLPP, 256), 1), 256, 0, stream>>>(
      Wm1, Wm1p, MLPP, MLPP, MLPP, MLP_HID, MLPP, MLP_HID, MLP_HID, 0L, 0L);
  k_stage_pad<<<dim3(ceil_div(MLPP * NOUT, 256), 1), 256, 0, stream>>>(
      Wm2, Wm2p, MLPP, NOUT, NOUT, MLP_HID, MLPP, NOUT, NOUT, 0L, 0L);
  k_stage_pad<<<dim3(ceil_div(MLPP, 256), 1), 256, 0, stream>>>(
      bm1, bm1p, 1, MLPP, MLPP, 1, 1, MLP_HID, MLP_HID, 0L, 0L);
  k_stage_pad<<<dim3(ceil_div(NOUT, 256), 1), 256, 0, stream>>>(
      bm2, bm2p, 1, NOUT, NOUT, 1, 1, NOUT, NOUT, 0L, 0L);

  const dim3 blk128(128);
  const dim3 gemmGridL(HIDP / 64, N / 64, H);        // 5 x 48 x 10
  const dim3 applyGridL(HIDP / 64, N / 64, H);

  // ================= Layer 0 =================
  k_wmma_gemm<<<gemmGridL, blk128, 0, stream>>>(
      x, W0p, bufA, F_IN_DIM, F_IN_DIM, HIDP, FEATP,
      0L, (long)F_IN_DIM * HIDP, (long)HIDP, nullptr, 0);
  k_compute_sd<<<ceil_div(N * H, 256), 256, 0, stream>>>(
      bufA, a0_src, a0_dst, sbuf, dbuf, N, H, HID_DIM, FEATP, HIDP);
  k_gat_stats<<<dim3(N, H), 32, 0, stream>>>(mask, sbuf, dbuf, rmaxb, rinvb, N);
  k_gat_apply<<<applyGridL, blk128, 0, stream>>>(
      bufA, mask, sbuf, dbuf, rmaxb, rinvb, bufB, N, FEATP, FEATP, (long)HIDP);

  // ================= Layer 1 =================
  k_wmma_gemm<<<gemmGridL, blk128, 0, stream>>>(
      bufB, W1p, bufA, FEATP, FEATP, HIDP, FEATP,
      0L, (long)FEATP * HIDP, (long)HIDP, nullptr, 0);
  k_compute_sd<<<ceil_div(N * H, 256), 256, 0, stream>>>(
      bufA, a1_src, a1_dst, sbuf, dbuf, N, H, HID_DIM, FEATP, HIDP);
  k_gat_stats<<<dim3(N, H), 32, 0, stream>>>(mask, sbuf, dbuf, rmaxb, rinvb, N);
  k_gat_apply<<<applyGridL, blk128, 0, stream>>>(
      bufA, mask, sbuf, dbuf, rmaxb, rinvb, bufC, N, FEATP, FEATP, (long)HIDP);

  // ================= Output attention layer (single head) =================
  k_wmma_gemm<<<dim3(OUTP / 64, N / 64, 1), blk128, 0, stream>>>(
      bufC, Wop, bufA, FEATP, FEATP, OUTP, OUTP, 0L, 0L, 0L, nullptr, 0);
  k_compute_sd<<<ceil_div(N, 256), 256, 0, stream>>>(
      bufA, ao_src, ao_dst, sbuf, dbuf, N, 1, OUT_ATT, OUTP, OUTP);
  k_gat_stats<<<dim3(N, 1), 32, 0, stream>>>(mask, sbuf, dbuf, rmaxb, rinvb, N);
  k_gat_apply<<<dim3(OUTP / 64, N / 64, 1), blk128, 0, stream>>>(
      bufA, mask, sbuf, dbuf, rmaxb, rinvb, bufB, N, OUTP, OUTP, 0L);

  // ================= Global mean pool (bufB, ld=OUTP) =================
  k_zero_f32<<<ceil_div(G_PAD * OUTP, 256), 256, 0, stream>>>(psum, (long)G_PAD * OUTP);
  k_zero_f32<<<ceil_div(G_PAD * OUTP, 256), 256, 0, stream>>>(pooled, (long)G_PAD * OUTP);
  k_zero_f32<<<1, 96, 0, stream>>>(pcnt, G);
  k_pool_accum<<<(unsigned)(((long)N * OUT_ATT + 255) / 256), 256, 0, stream>>>(
      bufB, batch, psum, pcnt, N, OUT_ATT, OUTP, OUTP);
  k_pool_norm<<<ceil_div(G * OUT_ATT, 256), 256, 0, stream>>>(
      psum, pcnt, pooled, G, OUT_ATT, OUTP);

  // ================= MLP head (padded rows, then copy out) =================
  k_wmma_gemm<<<dim3(MLPP / 64, G_PAD / 64, 1), blk128, 0, stream>>>(
      pooled, Wm1p, hidden, MLPP, OUTP, MLPP, MLPP, 0L, 0L, 0L, bm1p, 1);
  k_wmma_gemm<<<dim3(NOUT / 64, G_PAD / 64, 1), blk128, 0, stream>>>(
      hidden, Wm2p, outp, MLPP, MLPP, NOUT, NOUT, 0L, 0L, 0L, bm2p, 2);
  k_copy_out<<<ceil_div(G * NOUT, 256), 256, 0, stream>>>(outp, out, G, NOUT, NOUT);
}